// Comb1d_70471823393240
// MI455X (gfx1250) — compile-verified
//
#include <hip/hip_runtime.h>
#include <hip/hip_bf16.h>

#define NUM_TAPS     16
#define T_LEN        8192
#define O_TOTAL      64
#define B_TOTAL      8
#define O_PER_BLOCK  2
#define THREADS      256
#define PAD          4          // x[j] lives at xs[PAD + j]; keeps b128 LDS dst 16B-aligned

typedef __attribute__((address_space(3))) float lds_f;

__global__ __launch_bounds__(THREADS)
void comb1d_kernel(const float* __restrict__ x,
                   const float* __restrict__ f,
                   const float* __restrict__ a,
                   const int*   __restrict__ sr,
                   float*       __restrict__ y)
{
    // [0..3] = leading zeros, [4..T+3] = x row, [T+4..T+7] = trailing zeros
    __shared__ float xs[T_LEN + 2 * PAD];

    const int tid = threadIdx.x;
    const int blk = blockIdx.x;
    const int b   = blk / (O_TOTAL / O_PER_BLOCK);
    const int o0  = (blk % (O_TOTAL / O_PER_BLOCK)) * O_PER_BLOCK;

    // zero the pad slots (covers the dropped boundary masks)
    if (tid < 2 * PAD) {
        const int z = (tid < PAD) ? tid : (T_LEN + tid);
        xs[z] = 0.0f;
    }

    // ---- async bulk copy x[b, :] -> xs in LDS from slot PAD on (CDNA5 ASYNCcnt path, b128/lane) ----
    {
        const float4* gsrc = reinterpret_cast<const float4*>(x + (size_t)b * T_LEN);
        #pragma unroll
        for (int i = 0; i < (T_LEN / 4) / THREADS; ++i) {    // 8 x float4 per thread
            const int idx = tid + i * THREADS;
            const float4* gp = gsrc + idx;
            lds_f* lp = (lds_f*)(&xs[PAD + idx * 4]);        // byte addr 16 + 16*idx: aligned
            asm volatile("global_load_async_to_lds_b128 %0, %1, off"
                         :: "v"(lp), "v"(gp)
                         : "memory");
        }
        asm volatile("s_wait_asynccnt 0" ::: "memory");
    }
    __syncthreads();

    const float srf = (float)(*sr);

    for (int oo = 0; oo < O_PER_BLOCK; ++oo) {
        const int   o     = o0 + oo;
        const float delay = srf / f[o];
        const float av    = a[o];

        float dk[NUM_TAPS];
        float wk[NUM_TAPS];
        float w = 1.0f;
        #pragma unroll
        for (int k = 0; k < NUM_TAPS; ++k) {
            dk[k] = (float)k * delay;
            wk[k] = w;
            w *= av;
        }

        float* yrow = y + ((size_t)b * O_TOTAL + (size_t)o) * T_LEN;

        #pragma unroll 2
        for (int j = 0; j < T_LEN / THREADS; ++j) {          // 32 t per thread, coalesced
            const int   ti = tid + j * THREADS;
            const float t  = (float)ti;
            float acc = 0.0f;
            #pragma unroll
            for (int k = 0; k < NUM_TAPS; ++k) {
                const float pos  = t - dk[k];
                const float i0f  = floorf(pos);
                const float frac = pos - i0f;
                const int   i0   = (int)i0f;
                // med3(i0, -2, T-1) + PAD : pads make both boundary masks implicit
                const int   idx  = min(max(i0, -2), T_LEN - 1) + PAD;
                const float x0   = xs[idx];                  // merge -> ds_load_2addr_b32
                const float x1   = xs[idx + 1];
                const float w0   = 1.0f - frac;
                acc += wk[k] * (w0 * x0 + frac * x1);
            }
            __builtin_nontemporal_store(acc, &yrow[ti]);     // y is write-once: TH=NT
        }
    }
}

extern "C" void kernel_launch(void* const* d_in, const int* in_sizes, int n_in,
                              void* d_out, int out_size, void* d_ws, size_t ws_size,
                              hipStream_t stream)
{
    const float* x  = (const float*)d_in[0];   // (8, 1, 8192) f32
    const float* f  = (const float*)d_in[1];   // (64, 1) f32
    const float* a  = (const float*)d_in[2];   // (64, 1) f32
    const int*   sr = (const int*)  d_in[3];   // scalar int
    float*       y  = (float*)d_out;           // (8, 64, 8192) f32

    dim3 grid(B_TOTAL * (O_TOTAL / O_PER_BLOCK));   // 256 workgroups
    dim3 block(THREADS);                             // 8 waves (wave32)
    comb1d_kernel<<<grid, block, 0, stream>>>(x, f, a, sr, y);
}